// GNN_69707319214464
// MI455X (gfx1250) — compile-verified
//
#include <hip/hip_runtime.h>

typedef float f32x2 __attribute__((ext_vector_type(2)));
typedef float f32x4 __attribute__((ext_vector_type(4)));
typedef float f32x8 __attribute__((ext_vector_type(8)));

#define NND 50000
#define NED 800000
#define HDIM 128

// ---------------------------------------------------------------- stage A ---
__global__ void k_init_s(float* __restrict__ S, float* __restrict__ deg,
                         const int* __restrict__ sli, const int* __restrict__ slt) {
    int t = blockIdx.x * blockDim.x + threadIdx.x;
    if (t >= NND * 16) return;
    int k = t & 15;
    S[t] = (k == *sli) ? (float)(*slt) : 0.0f;
    if (k == 0) deg[t >> 4] = 1.0f;   // self loop counts once
}

__global__ void k_scat_s(const int* __restrict__ ei, const float* __restrict__ ea,
                         float* __restrict__ S, float* __restrict__ deg) {
    int t = blockIdx.x * blockDim.x + threadIdx.x;
    if (t >= NED * 16) return;
    int e = t >> 4, k = t & 15;
    int dst = ei[e];
    atomicAdd(&S[(size_t)dst * 16 + k], ea[t]);
    if (k == 0) atomicAdd(&deg[dst], 1.0f);
}

// Pack W[K,NC] into WMMA-B fragment order: P[(s*NCT+ct)*32+lane] = float2
// { W[4s+2*(lane>>4)][ct*16+(lane&15)], W[4s+2*(lane>>4)+1][same col] }
__global__ void k_pack(const float* __restrict__ W, float* __restrict__ P,
                       int K, int NC) {
    int t = blockIdx.x * blockDim.x + threadIdx.x;
    int total = (K >> 2) * (NC >> 4) * 32;
    if (t >= total) return;
    int lane = t & 31;
    int rest = t >> 5;
    int nct  = NC >> 4;
    int ct   = rest % nct;
    int s    = rest / nct;
    int kk   = 4 * s + 2 * (lane >> 4);
    int c    = ct * 16 + (lane & 15);
    P[2 * t]     = W[(size_t)kk * NC + c];
    P[2 * t + 1] = W[(size_t)(kk + 1) * NC + c];
}

__global__ void k_bnprep(const float* __restrict__ g0, const float* __restrict__ be0,
                         const float* __restrict__ rm0, const float* __restrict__ rv0,
                         const float* __restrict__ g,  const float* __restrict__ be,
                         const float* __restrict__ rm, const float* __restrict__ rv,
                         float* __restrict__ scale, float* __restrict__ shift) {
    int t = blockIdx.x * blockDim.x + threadIdx.x;
    if (t >= 3 * 256) return;
    int l = t >> 8, c = t & 255;
    float gg, bb, m, v;
    if (l == 0) { gg = g0[c]; bb = be0[c]; m = rm0[c]; v = rv0[c]; }
    else { int o = (l - 1) * 256 + c; gg = g[o]; bb = be[o]; m = rm[o]; v = rv[o]; }
    float sc = gg * rsqrtf(v + 1e-5f);
    scale[t] = sc;
    shift[t] = bb - m * sc;
}

// ---------------------------------------------------------------- per layer -
__global__ void k_selfloop_copy(const float* __restrict__ hin, float* __restrict__ agg) {
    int t = blockIdx.x * blockDim.x + threadIdx.x;
    if (t >= NND * (HDIM / 4)) return;
    *((f32x4*)agg + t) = *((const f32x4*)hin + t);
}

__global__ void k_scat_h(const int* __restrict__ ei, const float* __restrict__ hin,
                         float* __restrict__ agg) {
    int t = blockIdx.x * blockDim.x + threadIdx.x;
    if (t >= NED * 32) return;
    int e = t >> 5, q = t & 31;
    int dst = ei[e];
    int src = ei[NED + e];
    f32x4 v = *(const f32x4*)(hin + (size_t)src * HDIM + q * 4);
    float* p = agg + (size_t)dst * HDIM + q * 4;
    atomicAdd(p + 0, v.x);
    atomicAdd(p + 1, v.y);
    atomicAdd(p + 2, v.z);
    atomicAdd(p + 3, v.w);
}

// Fused per-node-tile layer kernel: one wave32 handles 16 rows.
// CONCAT / RELU_OUT are compile-time to keep the inner pipeline branch-free.
template <int CONCAT, int RELU_OUT>
__global__ __launch_bounds__(32)
void k_gin(const float* __restrict__ agg, const float* __restrict__ S,
           const float* __restrict__ deg, const float* __restrict__ encp,
           const float* __restrict__ encb, const float* __restrict__ w1p,
           const float* __restrict__ b1, const float* __restrict__ bnscale,
           const float* __restrict__ bnshift, const float* __restrict__ w2p,
           const float* __restrict__ b2, float* __restrict__ out)
{
    __shared__ float u[16 * 260];   // A operand of mid GEMM (padded stride)
    __shared__ float tt[16 * 260];  // A operand of out GEMM
    __shared__ float degs[16];

    const int lane = threadIdx.x;
    const int row0 = blockIdx.x * 16;
    const int lrow = lane & 15;
    const int hi   = lane >> 4;

    if (lane < 16) degs[lane] = deg[row0 + lane];

    // load aggregated-neighbor tile [16 x 128] into LDS
    for (int r = 0; r < 16; ++r) {
        f32x4 v = *(const f32x4*)(agg + (size_t)(row0 + r) * HDIM + lane * 4);
        int d = r * 260 + lane * 4;
        u[d + 0] = v.x; u[d + 1] = v.y; u[d + 2] = v.z; u[d + 3] = v.w;
    }
    __syncthreads();

    // A fragments of the encoder GEMM (K=16 -> 4 steps, kept in registers)
    f32x2 sa[4];
#pragma unroll
    for (int s = 0; s < 4; ++s)
        sa[s] = *(const f32x2*)(S + (size_t)(row0 + lrow) * 16 + 4 * s + 2 * hi);

    // ---- edge-encoder GEMM: E = S_tile[16x16] @ encW[16x128] + deg*encB ----
#pragma unroll
    for (int ct = 0; ct < 8; ++ct) {
        f32x8 acc = {0.f, 0.f, 0.f, 0.f, 0.f, 0.f, 0.f, 0.f};
#pragma unroll
        for (int s = 0; s < 4; ++s) {
            f32x2 b = *(const f32x2*)(encp + ((size_t)(s * 8 + ct) * 32 + lane) * 2);
            acc = __builtin_amdgcn_wmma_f32_16x16x4_f32(
                      false, sa[s], false, b, (short)0, acc, false, false);
        }
        int col = ct * 16 + lrow;
        float eb = encb[col];
#pragma unroll
        for (int r = 0; r < 8; ++r) {
            int m = r + 8 * hi;
            float v = acc[r] + degs[m] * eb;
            if (CONCAT) u[m * 260 + 128 + col] = v;   // concat: cols 128..255
            else        u[m * 260 + col] += v;        // add:    cols 0..127
        }
    }
    __syncthreads();

    // ---- mid GEMM: t = relu(bn(u @ w1 + b1)), output width 256 ----
    constexpr int KS1 = CONCAT ? 64 : 32;   // K/4
    for (int ct = 0; ct < 16; ++ct) {
        int col = ct * 16 + lrow;
        // prefetch next column tile's packed-weight stream into cache
        if (ct < 15)
            __builtin_prefetch(w1p + ((size_t)(ct + 1) * 32 + lane) * 2, 0, 1);
        float bias = b1[col];
        f32x8 acc = {bias, bias, bias, bias, bias, bias, bias, bias};
#pragma unroll 4
        for (int s = 0; s < KS1; ++s) {
            f32x2 a = *(const f32x2*)(&u[lrow * 260 + 4 * s + 2 * hi]);
            f32x2 b = *(const f32x2*)(w1p + ((size_t)(s * 16 + ct) * 32 + lane) * 2);
            acc = __builtin_amdgcn_wmma_f32_16x16x4_f32(
                      false, a, false, b, (short)0, acc, false, false);
        }
        float sc = bnscale[col], sh = bnshift[col];
#pragma unroll
        for (int r = 0; r < 8; ++r) {
            float v = acc[r] * sc + sh;
            v = v > 0.f ? v : 0.f;
            tt[(r + 8 * hi) * 260 + col] = v;
        }
    }
    __syncthreads();

    // ---- out GEMM: o = t @ w2 + b2, K=256, output width 128 ----
    for (int ct = 0; ct < 8; ++ct) {
        int col = ct * 16 + lrow;
        if (ct < 7)
            __builtin_prefetch(w2p + ((size_t)(ct + 1) * 32 + lane) * 2, 0, 1);
        float bias = b2[col];
        f32x8 acc = {bias, bias, bias, bias, bias, bias, bias, bias};
#pragma unroll 4
        for (int s = 0; s < 64; ++s) {
            f32x2 a = *(const f32x2*)(&tt[lrow * 260 + 4 * s + 2 * hi]);
            f32x2 b = *(const f32x2*)(w2p + ((size_t)(s * 8 + ct) * 32 + lane) * 2);
            acc = __builtin_amdgcn_wmma_f32_16x16x4_f32(
                      false, a, false, b, (short)0, acc, false, false);
        }
#pragma unroll
        for (int r = 0; r < 8; ++r) {
            float v = acc[r];
            if (RELU_OUT) v = v > 0.f ? v : 0.f;
            out[(size_t)(row0 + r + 8 * hi) * HDIM + col] = v;
        }
    }
}

// ---------------------------------------------------------------- host ------
extern "C" void kernel_launch(void* const* d_in, const int* in_sizes, int n_in,
                              void* d_out, int out_size, void* d_ws, size_t ws_size,
                              hipStream_t stream) {
    const float* x    = (const float*)d_in[0];
    const int*   ei   = (const int*)d_in[1];
    const float* ea   = (const float*)d_in[2];
    const int*   sli  = (const int*)d_in[3];
    const int*   slt  = (const int*)d_in[4];
    const float* encw0 = (const float*)d_in[5];
    const float* encb0 = (const float*)d_in[6];
    const float* w1_0  = (const float*)d_in[7];
    const float* b1_0  = (const float*)d_in[8];
    const float* g0    = (const float*)d_in[9];
    const float* be0   = (const float*)d_in[10];
    const float* rm0   = (const float*)d_in[11];
    const float* rv0   = (const float*)d_in[12];
    const float* w2_0  = (const float*)d_in[13];
    const float* b2_0  = (const float*)d_in[14];
    const float* encw  = (const float*)d_in[15];
    const float* encb  = (const float*)d_in[16];
    const float* w1    = (const float*)d_in[17];
    const float* b1    = (const float*)d_in[18];
    const float* g     = (const float*)d_in[19];
    const float* be    = (const float*)d_in[20];
    const float* rm    = (const float*)d_in[21];
    const float* rv    = (const float*)d_in[22];
    const float* w2    = (const float*)d_in[23];
    const float* b2    = (const float*)d_in[24];

    float* ws = (float*)d_ws;
    size_t off = 0;
    auto take = [&](size_t n) { float* p = ws + off; off += (n + 255) & ~(size_t)255; return p; };

    float* S    = take((size_t)NND * 16);
    float* deg  = take(NND);
    float* agg  = take((size_t)NND * HDIM);
    float* hA   = take((size_t)NND * HDIM);
    float* hB   = take((size_t)NND * HDIM);
    float* encp0 = take(16 * 128);
    float* encp1 = take(16 * 128);
    float* encp2 = take(16 * 128);
    float* w1p0  = take(128 * 256);
    float* w1p1  = take(256 * 256);
    float* w1p2  = take(256 * 256);
    float* w2p0  = take(256 * 128);
    float* w2p1  = take(256 * 128);
    float* w2p2  = take(256 * 128);
    float* bnsc  = take(3 * 256);
    float* bnsh  = take(3 * 256);

    const int TB = 256;
    auto blocks = [](long n, int tb) { return (unsigned)((n + tb - 1) / tb); };

    // ---- once-per-launch prep ----
    k_init_s<<<blocks((long)NND * 16, TB), TB, 0, stream>>>(S, deg, sli, slt);
    k_scat_s<<<blocks((long)NED * 16, TB), TB, 0, stream>>>(ei, ea, S, deg);
    k_bnprep<<<blocks(3 * 256, TB), TB, 0, stream>>>(g0, be0, rm0, rv0, g, be, rm, rv, bnsc, bnsh);

    auto pack = [&](const float* W, float* P, int K, int NC) {
        long total = (long)(K / 4) * (NC / 16) * 32;
        k_pack<<<blocks(total, TB), TB, 0, stream>>>(W, P, K, NC);
    };
    pack(encw0, encp0, 16, 128);
    pack(encw + 0 * 16 * 128, encp1, 16, 128);
    pack(encw + 1 * 16 * 128, encp2, 16, 128);
    pack(w1_0, w1p0, 128, 256);
    pack(w1 + 0 * 256 * 256, w1p1, 256, 256);
    pack(w1 + 1 * 256 * 256, w1p2, 256, 256);
    pack(w2_0, w2p0, 256, 128);
    pack(w2 + 0 * 256 * 128, w2p1, 256, 128);
    pack(w2 + 1 * 256 * 128, w2p2, 256, 128);

    // ---- layer 0: concat=0, relu_out=1 ----
    k_selfloop_copy<<<blocks((long)NND * (HDIM / 4), TB), TB, 0, stream>>>(x, agg);
    k_scat_h<<<blocks((long)NED * 32, TB), TB, 0, stream>>>(ei, x, agg);
    k_gin<0, 1><<<NND / 16, 32, 0, stream>>>(agg, S, deg, encp0, encb0,
                                             w1p0, b1_0, bnsc + 0, bnsh + 0,
                                             w2p0, b2_0, hA);

    // ---- layer 1: concat=1, relu_out=1 ----
    k_selfloop_copy<<<blocks((long)NND * (HDIM / 4), TB), TB, 0, stream>>>(hA, agg);
    k_scat_h<<<blocks((long)NED * 32, TB), TB, 0, stream>>>(ei, hA, agg);
    k_gin<1, 1><<<NND / 16, 32, 0, stream>>>(agg, S, deg, encp1, encb,
                                             w1p1, b1, bnsc + 256, bnsh + 256,
                                             w2p1, b2, hB);

    // ---- layer 2: concat=1, relu_out=0 ----
    k_selfloop_copy<<<blocks((long)NND * (HDIM / 4), TB), TB, 0, stream>>>(hB, agg);
    k_scat_h<<<blocks((long)NED * 32, TB), TB, 0, stream>>>(ei, hB, agg);
    k_gin<1, 0><<<NND / 16, 32, 0, stream>>>(agg, S, deg, encp2, encb + 128,
                                             w1p2, b1 + 256, bnsc + 512, bnsh + 512,
                                             w2p2, b2 + 128, (float*)d_out);
}